// LAV_42528766165258
// MI455X (gfx1250) — compile-verified
//
#include <hip/hip_runtime.h>
#include <hip/hip_bf16.h>

typedef __attribute__((ext_vector_type(2))) float v2f;
typedef __attribute__((ext_vector_type(8))) float v8f;

#define NSEQ 2048
#define DIM  128
#define BIGF 1e10f
#define GAMMA_F 0.1f

// ---------------------------------------------------------------------------
// Row squared-norms for X (2048x128) and Y (2048x128): one wave32 per row.
// ---------------------------------------------------------------------------
__global__ void __launch_bounds__(128)
norms_kernel(const float* __restrict__ X, const float* __restrict__ Y,
             float* __restrict__ xn, float* __restrict__ yn) {
  const int lane  = threadIdx.x & 31;
  const int wave  = (blockIdx.x * blockDim.x + threadIdx.x) >> 5;
  const int nwave = (gridDim.x * blockDim.x) >> 5;
  for (int row = wave; row < 2 * NSEQ; row += nwave) {
    const float* src = (row < NSEQ) ? (X + (size_t)row * DIM)
                                    : (Y + (size_t)(row - NSEQ) * DIM);
    float s = 0.f;
    #pragma unroll
    for (int c = 0; c < DIM; c += 32) { float v = src[c + lane]; s += v * v; }
    #pragma unroll
    for (int off = 16; off > 0; off >>= 1) s += __shfl_xor(s, off, 32);
    if (lane == 0) { if (row < NSEQ) xn[row] = s; else yn[row - NSEQ] = s; }
  }
}

// ---------------------------------------------------------------------------
// Pairwise squared distances via V_WMMA_F32_16X16X4_F32.
// Each wave32 owns one 16x16 output tile; 8 waves per block.
// D[r][c] = max(0, |x_r|^2 + |y_c|^2 - 2 * x_r . y_c)
// A fragment (16x4 f32): lane&15 = M row, (lane>>4)*2 = K sub-offset -> one b64 load.
// B fragment (4x16 f32): B = Y^T, so same addressing with rows of Y.
// ---------------------------------------------------------------------------
__global__ void __launch_bounds__(256)
dist_kernel(const float* __restrict__ X, const float* __restrict__ Y,
            const float* __restrict__ xn, const float* __restrict__ yn,
            float* __restrict__ Dout) {
  const int lane   = threadIdx.x & 31;
  const int waveId = threadIdx.x >> 5;
  const int tile   = blockIdx.x * 8 + waveId;      // 128x128 tiles
  const int row0   = (tile >> 7) * 16;
  const int col0   = (tile & 127) * 16;
  const int m      = lane & 15;
  const int koff   = (lane >> 4) * 2;

  const float* arow = X + (size_t)(row0 + m) * DIM;
  const float* brow = Y + (size_t)(col0 + m) * DIM;

  v8f c = {};
  #pragma unroll 4
  for (int k = 0; k < DIM; k += 4) {
    v2f a = *(const v2f*)(arow + k + koff);
    v2f b = *(const v2f*)(brow + k + koff);
    c = __builtin_amdgcn_wmma_f32_16x16x4_f32(false, a, false, b,
                                              (short)0, c, false, false);
  }

  const int ncol  = col0 + (lane & 15);
  const int mbase = row0 + ((lane >> 4) << 3);
  const float ynv = yn[ncol];
  #pragma unroll
  for (int v = 0; v < 8; ++v) {
    const int row = mbase + v;
    float d = xn[row] + ynv - 2.0f * c[v];
    Dout[(size_t)row * NSEQ + ncol] = fmaxf(d, 0.0f);
  }
}

// ---------------------------------------------------------------------------
// Soft-DTW wavefront DP. One 1024-thread block (32 wave32s) per matrix;
// each thread owns rows i and i+1024 of the anti-diagonal.
// Triple-buffered rows in LDS (24 KB of the WGP's 320 KB): step k writes the
// buffer holding r^{k-3}, which no thread reads at step k, so a SINGLE
// __syncthreads per step suffices (vs 2 for ping-pong).
// Per thread, D addresses advance sequentially along its row; prefetch the
// next 128B cacheline once per 32 steps (global_prefetch_b8, counter-free).
// ---------------------------------------------------------------------------
__global__ void __launch_bounds__(1024)
softdtw_kernel(const float* __restrict__ Dbase, long long strideElems,
               float* __restrict__ outs) {
  __shared__ float R[3][NSEQ];
  const float* D = Dbase + (long long)blockIdx.x * strideElems;
  const int tid = threadIdx.x;
  const float g = GAMMA_F, invg = 1.0f / GAMMA_F;

  // r^{0}: BIG except r[0] = D[0][0]  -> buffer 0
  // r^{-1}: BIG everywhere            -> buffer 2
  R[0][tid]        = (tid == 0) ? D[0] : BIGF;
  R[0][tid + 1024] = BIGF;
  R[2][tid]        = BIGF;
  R[2][tid + 1024] = BIGF;
  __syncthreads();

  int r1b = 0;   // buffer holding r^{k-1}
  int r2b = 2;   // buffer holding r^{k-2}
  int wb  = 1;   // buffer to write r^{k}

  const int K = 2 * NSEQ - 1;
  for (int k = 1; k < K; ++k) {
    float rnew[2];
    #pragma unroll
    for (int t = 0; t < 2; ++t) {
      const int i = tid + t * 1024;
      const int j = k - i;
      if ((k & 31) == 0) {
        // next cacheline of this thread's D row (always a valid address)
        int pj = j + 33; pj = pj < 0 ? 0 : (pj > NSEQ - 1 ? NSEQ - 1 : pj);
        __builtin_prefetch(D + (size_t)i * NSEQ + pj, 0, 1);
      }
      const float d    = (j >= 0 && j < NSEQ) ? D[(size_t)i * NSEQ + j] : BIGF;
      const float diag = (i > 0) ? R[r2b][i - 1] : BIGF;
      const float up   = (i > 0) ? R[r1b][i - 1] : BIGF;
      const float left = R[r1b][i];
      // smin = -g * logsumexp(-diag/g, -up/g, -left/g), max-subtracted form
      const float mn = fminf(fminf(diag, up), left);
      const float s  = __expf((mn - diag) * invg) +
                       __expf((mn - up)   * invg) +
                       __expf((mn - left) * invg);
      rnew[t] = d + mn - g * __logf(s);
    }
    R[wb][tid]        = rnew[0];
    R[wb][tid + 1024] = rnew[1];
    __syncthreads();                 // single barrier: publish r^{k}
    const int tmp = r2b; r2b = r1b; r1b = wb; wb = tmp;
  }
  if (tid == 0) outs[blockIdx.x] = R[r1b][NSEQ - 1];
}

// ---------------------------------------------------------------------------
// Contrastive IDM partial reduction (deterministic two-stage tree).
// ---------------------------------------------------------------------------
__global__ void __launch_bounds__(256)
idm_kernel(const float* __restrict__ Dm, const float* __restrict__ idx,
           const int* __restrict__ lenp, float* __restrict__ partials) {
  __shared__ float red[256];
  const int tid = threadIdx.x;
  const float thr = 10.0f / (float)lenp[0];    // SIGMA / seq_len
  float s = 0.f;
  const size_t total = (size_t)NSEQ * NSEQ;
  for (size_t e = (size_t)blockIdx.x * 256 + tid; e < total;
       e += (size_t)gridDim.x * 256) {
    const int i = (int)(e >> 11);
    const int j = (int)(e & (NSEQ - 1));
    const float dist = Dm[e];
    const float gd   = idx[i] - idx[j];
    const float prob = fmaxf(2.0f - dist, 0.0f);   // relu(MARGIN - dist)
    const float w    = 1.0f + gd * gd;
    const bool  neg  = (fabsf(gd) - thr) > 0.0f;
    s += neg ? w * prob : dist;
  }
  red[tid] = s; __syncthreads();
  for (int off = 128; off > 0; off >>= 1) {
    if (tid < off) red[tid] += red[tid + off];
    __syncthreads();
  }
  if (tid == 0) partials[blockIdx.x] = red[0];
}

__global__ void __launch_bounds__(256)
reduce_kernel(const float* __restrict__ partials, int n, float* __restrict__ out) {
  __shared__ float red[256];
  const int tid = threadIdx.x;
  float s = 0.f;
  for (int i = tid; i < n; i += 256) s += partials[i];
  red[tid] = s; __syncthreads();
  for (int off = 128; off > 0; off >>= 1) {
    if (tid < off) red[tid] += red[tid + off];
    __syncthreads();
  }
  if (tid == 0) *out = red[0];
}

// ---------------------------------------------------------------------------
// total = (sdtw_xy - 0.5*(sdtw_xx + sdtw_yy) + idm_a + idm_b) / NUM_FRAMES
// ---------------------------------------------------------------------------
__global__ void combine_kernel(const float* __restrict__ sc, float* __restrict__ out) {
  if (threadIdx.x == 0 && blockIdx.x == 0) {
    const float pos = sc[0] - 0.5f * (sc[1] + sc[2]);
    out[0] = (pos + sc[3] + sc[4]) / 2048.0f;
  }
}

// ---------------------------------------------------------------------------
extern "C" void kernel_launch(void* const* d_in, const int* in_sizes, int n_in,
                              void* d_out, int out_size, void* d_ws, size_t ws_size,
                              hipStream_t stream) {
  (void)in_sizes; (void)n_in; (void)out_size;
  const float* X     = (const float*)d_in[0];   // a_emb (1,2048,128) f32
  const float* Y     = (const float*)d_in[1];   // b_emb
  const float* a_idx = (const float*)d_in[2];   // (2048,) f32
  const float* b_idx = (const float*)d_in[3];
  const int*   a_len = (const int*)d_in[4];     // scalar int
  const int*   b_len = (const int*)d_in[5];
  float* out = (float*)d_out;

  const size_t DBYTES    = (size_t)NSEQ * NSEQ * sizeof(float);
  const size_t tailBytes = (size_t)(NSEQ + NSEQ + 1024 + 8) * sizeof(float);
  char* ws = (char*)d_ws;

  const bool parallel = ws_size >= 3 * DBYTES + tailBytes;  // constant per run
  const size_t tailOff = (parallel ? 3 : 1) * DBYTES;
  float* xn       = (float*)(ws + tailOff);
  float* yn       = xn + NSEQ;
  float* partials = yn + NSEQ;
  float* sc       = partials + 1024;  // [sdtw_xy, sdtw_xx, sdtw_yy, idm_a, idm_b]

  norms_kernel<<<32, 128, 0, stream>>>(X, Y, xn, yn);

  float* D0 = (float*)ws;
  const int distGrid = (128 * 128) / 8;   // 8 waves (tiles) per 256-thread block

  if (parallel) {
    float* D1 = (float*)(ws + DBYTES);
    float* D2 = (float*)(ws + 2 * DBYTES);
    dist_kernel<<<distGrid, 256, 0, stream>>>(X, Y, xn, yn, D0);
    dist_kernel<<<distGrid, 256, 0, stream>>>(X, X, xn, xn, D1);
    dist_kernel<<<distGrid, 256, 0, stream>>>(Y, Y, yn, yn, D2);
    // three serial DPs run concurrently on three WGPs
    softdtw_kernel<<<3, 1024, 0, stream>>>(D0, (long long)NSEQ * NSEQ, sc);
    idm_kernel<<<1024, 256, 0, stream>>>(D1, a_idx, a_len, partials);
    reduce_kernel<<<1, 256, 0, stream>>>(partials, 1024, sc + 3);
    idm_kernel<<<1024, 256, 0, stream>>>(D2, b_idx, b_len, partials);
    reduce_kernel<<<1, 256, 0, stream>>>(partials, 1024, sc + 4);
  } else {
    dist_kernel<<<distGrid, 256, 0, stream>>>(X, Y, xn, yn, D0);
    softdtw_kernel<<<1, 1024, 0, stream>>>(D0, 0, sc + 0);
    dist_kernel<<<distGrid, 256, 0, stream>>>(X, X, xn, xn, D0);
    idm_kernel<<<1024, 256, 0, stream>>>(D0, a_idx, a_len, partials);
    reduce_kernel<<<1, 256, 0, stream>>>(partials, 1024, sc + 3);
    softdtw_kernel<<<1, 1024, 0, stream>>>(D0, 0, sc + 1);
    dist_kernel<<<distGrid, 256, 0, stream>>>(Y, Y, yn, yn, D0);
    idm_kernel<<<1024, 256, 0, stream>>>(D0, b_idx, b_len, partials);
    reduce_kernel<<<1, 256, 0, stream>>>(partials, 1024, sc + 4);
    softdtw_kernel<<<1, 1024, 0, stream>>>(D0, 0, sc + 2);
  }
  combine_kernel<<<1, 1, 0, stream>>>(sc, out);
}